// GraphConv_88106959110341
// MI455X (gfx1250) — compile-verified
//
#include <hip/hip_runtime.h>

// GraphConv scatter-add:  out[tidx[e]] += input[sidx[e]] * (esgn[e]*enorm[e])
// N_NODES=100000, N_FEAT=128, N_EDGES=600000.
// Memory/atomic bound; both input (51.2MB) and output (51.2MB) are L2-resident
// on MI455X (192MB L2): gathers hit L2, scatter uses native L2 f32 atomics.
// CDNA5-specific paths: global_load_async_to_lds_b32 (ASYNCcnt) metadata
// staging, wave32 row layout (32 lanes x float4 = 128-float row),
// fire-and-forget global_atomic_add_f32.

#define FEAT     128
#define EPB      64     // edges per block (64 edges x 4 dwords meta = 256 dwords)
#define THREADS  256    // 8 waves of 32 (wave32)
#define EPW      (EPB / 8)   // edges per wave = 8

typedef __attribute__((address_space(1))) int* gint_p;   // global int*
typedef __attribute__((address_space(3))) int* lint_p;   // LDS int*

// Fire-and-forget native f32 atomic add (no CAS loop, no return; STOREcnt-tracked).
__device__ __forceinline__ void atomic_add_f32_nort(float* p, float v) {
    asm volatile("global_atomic_add_f32 %0, %1, off" :: "v"(p), "v"(v) : "memory");
}

__global__ __launch_bounds__(256) void zero_f4(float4* __restrict__ out, int n4) {
    int i = blockIdx.x * 256 + threadIdx.x;
    if (i < n4) out[i] = make_float4(0.f, 0.f, 0.f, 0.f);
}

__global__ __launch_bounds__(THREADS) void gconv_scatter(
    const float* __restrict__ in,
    const int*   __restrict__ sidx,
    const int*   __restrict__ tidx,
    const float* __restrict__ enorm,
    const float* __restrict__ esgn,
    float*       __restrict__ out,
    int n_edges)
{
    // Layout: [0,64) sidx | [64,128) tidx | [128,192) enorm bits | [192,256) esgn bits
    __shared__ int meta[4 * EPB];

    const int tid  = threadIdx.x;
    const int base = blockIdx.x * EPB;

    // ---- Stage 64 edges' metadata into LDS, one dword per thread (async path) ----
    {
        const int a = tid >> 6;            // which of the 4 arrays
        int i = tid & (EPB - 1);
        if (base + i >= n_edges) i = n_edges - 1 - base;   // clamp for last block
        const int* srcs[4] = { sidx, tidx, (const int*)enorm, (const int*)esgn };
        const int* g = srcs[a] + base + i;
#if defined(__has_builtin) && __has_builtin(__builtin_amdgcn_global_load_async_to_lds_b32)
        __builtin_amdgcn_global_load_async_to_lds_b32(
            (gint_p)(int*)g, (lint_p)&meta[tid], 0, 0);
  #if __has_builtin(__builtin_amdgcn_s_wait_asynccnt)
        __builtin_amdgcn_s_wait_asynccnt(0);
  #else
        asm volatile("s_wait_asynccnt 0" ::: "memory");
  #endif
#else
        meta[tid] = *g;
#endif
    }
    __syncthreads();

    const int wave = tid >> 5;   // 0..7
    const int lane = tid & 31;   // 32 lanes x float4 = one 128-float row

    if (base + EPB <= n_edges) {
        // ---- Fast path: full block, guard-free, explicitly pipelined ----
        float4 v[EPW];
        float  w[EPW];
        int    t[EPW];

        // Phase 1: issue all 8 row gathers back-to-back (8 b128 loads in flight).
        #pragma unroll
        for (int k = 0; k < EPW; ++k) {
            const int e = wave + (k << 3);          // wave-uniform
            const int s = meta[e];
            t[k] = meta[EPB + e];
            w[k] = __int_as_float(meta[2 * EPB + e]) *
                   __int_as_float(meta[3 * EPB + e]);
            v[k] = ((const float4*)(in + (size_t)s * FEAT))[lane];
        }

        // Phase 2: fire 32 atomics; loads above already issued/landed.
        #pragma unroll
        for (int k = 0; k < EPW; ++k) {
            float* dst = out + (size_t)t[k] * FEAT + (lane << 2);
            atomic_add_f32_nort(dst + 0, v[k].x * w[k]);
            atomic_add_f32_nort(dst + 1, v[k].y * w[k]);
            atomic_add_f32_nort(dst + 2, v[k].z * w[k]);
            atomic_add_f32_nort(dst + 3, v[k].w * w[k]);
        }
    } else {
        // ---- Tail path: guarded per edge ----
        #pragma unroll
        for (int k = 0; k < EPW; ++k) {
            const int e = wave + (k << 3);
            if (base + e >= n_edges) continue;
            const int   s = meta[e];
            const int   t = meta[EPB + e];
            const float w = __int_as_float(meta[2 * EPB + e]) *
                            __int_as_float(meta[3 * EPB + e]);
            const float4 v = ((const float4*)(in + (size_t)s * FEAT))[lane];
            float* dst = out + (size_t)t * FEAT + (lane << 2);
            atomic_add_f32_nort(dst + 0, v.x * w);
            atomic_add_f32_nort(dst + 1, v.y * w);
            atomic_add_f32_nort(dst + 2, v.z * w);
            atomic_add_f32_nort(dst + 3, v.w * w);
        }
    }
}

extern "C" void kernel_launch(void* const* d_in, const int* in_sizes, int n_in,
                              void* d_out, int out_size, void* d_ws, size_t ws_size,
                              hipStream_t stream) {
    const float* in    = (const float*)d_in[0];
    const int*   sidx  = (const int*)  d_in[1];
    const int*   tidx  = (const int*)  d_in[2];
    const float* enorm = (const float*)d_in[3];
    const float* esgn  = (const float*)d_in[4];
    float*       out   = (float*)d_out;

    const int n_edges = in_sizes[1];

    // Zero the accumulator (harness poisons d_out).
    const int n4 = out_size / 4;
    zero_f4<<<(n4 + 255) / 256, 256, 0, stream>>>((float4*)out, n4);

    // Edge scatter with L2-resident f32 atomics.
    const int blocks = (n_edges + EPB - 1) / EPB;
    gconv_scatter<<<blocks, THREADS, 0, stream>>>(in, sidx, tidx, enorm, esgn, out, n_edges);
}